// TotalRandom2D_4483945857084
// MI455X (gfx1250) — compile-verified
//
#include <hip/hip_runtime.h>

// Pure data-movement kernel: out[b,c,i,j] = x[b,c,2i+ki, 2j+kj],
// ki/kj from idx[b,i,j] in {0..3}.  HBM-roofline bound (~1.34 GB @ 23.3 TB/s).

typedef float v4f __attribute__((ext_vector_type(4)));
typedef int   v4i __attribute__((ext_vector_type(4)));

constexpr int kB  = 16;
constexpr int kC  = 64;
constexpr int kH  = 512;
constexpr int kW  = 512;
constexpr int kHO = 256;   // (512-2)/2 + 1
constexpr int kWO = 256;
constexpr int kCLANES = 4;             // channels handled in parallel per block
constexpr int kCITERS = kC / kCLANES;  // 16 channel-loop iterations per thread

__device__ __forceinline__ float sel4(bool ki, bool kj,
                                      float r0e, float r0o,
                                      float r1e, float r1o) {
    float a = kj ? r0o : r0e;   // v_cndmask
    float b = kj ? r1o : r1e;   // v_cndmask
    return ki ? b : a;          // v_cndmask
}

__global__ __launch_bounds__(256)
void TotalRandom2D_gather_kernel(const float* __restrict__ x,
                                 const int*   __restrict__ idx,
                                 float*       __restrict__ out) {
    const int tid   = threadIdx.x;
    const int j4    = tid & 63;        // 64 groups of 4 output columns
    const int cLane = tid >> 6;        // 0..3
    const int bi    = blockIdx.x;      // 0 .. kB*kHO-1
    const int b     = bi >> 8;         // / kHO (kHO == 256)
    const int i     = bi & 255;        // % kHO

    // --- idx for this thread's 4 output columns; reused for all 16 channels of this thread ---
    const v4i id4 = *(const v4i*)(idx + ((long)b * kHO + i) * kWO + 4 * j4);
    const bool ki0 = (id4.x & 2), kj0 = (id4.x & 1);
    const bool ki1 = (id4.y & 2), kj1 = (id4.y & 1);
    const bool ki2 = (id4.z & 2), kj2 = (id4.z & 1);
    const bool ki3 = (id4.w & 2), kj3 = (id4.w & 1);

    // --- base pointers for the first channel this thread owns ---
    const long c0 = (long)cLane * kCITERS;                 // contiguous channel range
    const float* xp = x   + (((long)(b * kC) + c0) * kH + 2 * i) * kW + 8 * j4;
    float*       op = out + (((long)(b * kC) + c0) * kHO + i) * kWO + 4 * j4;

    constexpr long xChanStride = (long)kH * kW;    // 1 channel of x
    constexpr long oChanStride = (long)kHO * kWO;  // 1 channel of out

#pragma unroll 4
    for (int it = 0; it < kCITERS; ++it) {
        // Two full input rows covering x columns [8*j4, 8*j4+8): dense,
        // coalesced 128-bit streaming loads; every byte used exactly once
        // across the grid -> non-temporal.
        v4f r0a = __builtin_nontemporal_load((const v4f*)(xp));
        v4f r0b = __builtin_nontemporal_load((const v4f*)(xp + 4));
        v4f r1a = __builtin_nontemporal_load((const v4f*)(xp + kW));
        v4f r1b = __builtin_nontemporal_load((const v4f*)(xp + kW + 4));

        // Keep the HBM pipe primed across the 1 MB channel-stride jump.
        // global_prefetch_b8: no LOADcnt cost, speculative RT.
        if (it + 1 < kCITERS) {
            __builtin_prefetch(xp + xChanStride, 0, 0);
            __builtin_prefetch(xp + xChanStride + kW, 0, 0);
        }

        v4f r;
        r.x = sel4(ki0, kj0, r0a.x, r0a.y, r1a.x, r1a.y);
        r.y = sel4(ki1, kj1, r0a.z, r0a.w, r1a.z, r1a.w);
        r.z = sel4(ki2, kj2, r0b.x, r0b.y, r1b.x, r1b.y);
        r.w = sel4(ki3, kj3, r0b.z, r0b.w, r1b.z, r1b.w);

        __builtin_nontemporal_store(r, (v4f*)op);

        xp += xChanStride;
        op += oChanStride;
    }
}

extern "C" void kernel_launch(void* const* d_in, const int* in_sizes, int n_in,
                              void* d_out, int out_size, void* d_ws, size_t ws_size,
                              hipStream_t stream) {
    const float* x   = (const float*)d_in[0];
    const int*   idx = (const int*)d_in[1];
    float*       out = (float*)d_out;

    // One block per (b, i) output row; 8 wave32 per block; 4096 blocks.
    dim3 grid(kB * kHO);
    dim3 block(256);
    TotalRandom2D_gather_kernel<<<grid, block, 0, stream>>>(x, idx, out);
}